// TriadSwinEncoder_35673998360603
// MI455X (gfx1250) — compile-verified
//
#include <hip/hip_runtime.h>
#include <hip/hip_bf16.h>
#include <math.h>
#include <stdint.h>

// ---------------------------------------------------------------------------
// fp32 WMMA helpers (CDNA5 / gfx1250, wave32)
// ---------------------------------------------------------------------------
typedef __attribute__((ext_vector_type(2))) float v2f;
typedef __attribute__((ext_vector_type(8))) float v8f;

#define EPSN 1e-5f

__device__ __forceinline__ v8f wmma4(v2f a, v2f b, v8f c) {
  // D(16x16,f32) += A(16x4,f32) * B(4x16,f32); 8-arg VOP3P form
  return __builtin_amdgcn_wmma_f32_16x16x4_f32(false, a, false, b, (short)0, c,
                                               false, false);
}

__device__ __forceinline__ float wred(float v) {
#pragma unroll
  for (int o = 16; o > 0; o >>= 1) v += __shfl_xor(v, o, 32);
  return v;
}

__device__ __forceinline__ float lrelu_f(float v) { return v >= 0.f ? v : 0.01f * v; }
__device__ __forceinline__ float gelu_f(float v) {
  return 0.5f * v * (1.f + erff(v * 0.70710678118654752f));
}
// shifted-window region label (only used when ws=7, shift=3)
__device__ __forceinline__ int regio7(int u, int dp) {
  return (u < dp - 7) ? 0 : ((u < dp - 3) ? 1 : 2);
}

// ---------------------------------------------------------------------------
// Patch embed: conv 2x2x2 stride2 VALID, 1->48 ch, output channels-last
// ---------------------------------------------------------------------------
__global__ void pe_kernel(const float* __restrict__ x, const float* __restrict__ w,
                          const float* __restrict__ bias, float* __restrict__ out,
                          int total) {
  int idx = blockIdx.x * 256 + threadIdx.x;
  if (idx >= total) return;
  int oc = idx % 48;
  int vox = idx / 48;
  int xx = vox % 32, yy = (vox / 32) % 32, zz = (vox / 1024) % 32, b = vox / 32768;
  const float* xp = x + (((size_t)b * 64 + zz * 2) * 64 + yy * 2) * 64 + xx * 2;
  float acc = bias[oc];
#pragma unroll
  for (int i = 0; i < 2; i++)
#pragma unroll
    for (int j = 0; j < 2; j++)
#pragma unroll
      for (int k = 0; k < 2; k++)
        acc += xp[(size_t)i * 4096 + j * 64 + k] * w[oc * 8 + i * 4 + j * 2 + k];
  out[(size_t)vox * 48 + oc] = acc;
}

// ---------------------------------------------------------------------------
// proj_out: non-affine channel LayerNorm; channels-last in -> channels-first out
// ---------------------------------------------------------------------------
__global__ void proj_out_kernel(const float* __restrict__ in, float* __restrict__ out,
                                int nvox, int C, int sp) {
  int wid = (blockIdx.x * blockDim.x + threadIdx.x) >> 5;
  int lane = threadIdx.x & 31;
  if (wid >= nvox) return;
  const float* p = in + (size_t)wid * C;
  float s = 0.f, s2 = 0.f;
  for (int c = lane; c < C; c += 32) { float v = p[c]; s += v; s2 += v * v; }
  s = wred(s); s2 = wred(s2);
  float m = s / C;
  float r = rsqrtf(s2 / C - m * m + EPSN);
  int b = wid / sp, off = wid % sp;
  float* ob = out + (size_t)b * C * sp + off;
  for (int c = lane; c < C; c += 32) ob[(size_t)c * sp] = (p[c] - m) * r;
}

// ---------------------------------------------------------------------------
// token LayerNorm with affine (norm1 / norm2)
// ---------------------------------------------------------------------------
__global__ void ln_kernel(const float* __restrict__ in, float* __restrict__ out,
                          const float* __restrict__ g, const float* __restrict__ bb,
                          int ntok, int C) {
  int wid = (blockIdx.x * blockDim.x + threadIdx.x) >> 5;
  int lane = threadIdx.x & 31;
  if (wid >= ntok) return;
  const float* p = in + (size_t)wid * C;
  float s = 0.f, s2 = 0.f;
  for (int c = lane; c < C; c += 32) { float v = p[c]; s += v; s2 += v * v; }
  s = wred(s); s2 = wred(s2);
  float m = s / C;
  float r = rsqrtf(s2 / C - m * m + EPSN);
  float* q = out + (size_t)wid * C;
  for (int c = lane; c < C; c += 32) q[c] = (p[c] - m) * r * g[c] + bb[c];
}

// ---------------------------------------------------------------------------
// InstanceNorm stats (mean, rstd) per (b,c); channels-last input
// ---------------------------------------------------------------------------
__global__ void inorm_stats_kernel(const float* __restrict__ in, float* __restrict__ stats,
                                   int C, int S) {
  __shared__ float sh0[256], sh1[256];
  int bc = blockIdx.x;
  int b = bc / C, c = bc % C;
  float s = 0.f, s2 = 0.f;
  for (int i = threadIdx.x; i < S; i += 256) {
    float v = in[((size_t)b * S + i) * C + c];
    s += v; s2 += v * v;
  }
  sh0[threadIdx.x] = s; sh1[threadIdx.x] = s2;
  __syncthreads();
  for (int o = 128; o > 0; o >>= 1) {
    if (threadIdx.x < o) { sh0[threadIdx.x] += sh0[threadIdx.x + o]; sh1[threadIdx.x] += sh1[threadIdx.x + o]; }
    __syncthreads();
  }
  if (threadIdx.x == 0) {
    float m = sh0[0] / S;
    float var = sh1[0] / S - m * m;
    stats[bc * 2] = m;
    stats[bc * 2 + 1] = rsqrtf(var + EPSN);
  }
}

// mode 0: out = lrelu(norm(in));  mode 1: out = lrelu(norm(in) + skip)
__global__ void inorm_apply_kernel(const float* __restrict__ in, const float* __restrict__ stats,
                                   float* __restrict__ out, const float* __restrict__ skip,
                                   int total, int C, int S, int mode) {
  int idx = blockIdx.x * 256 + threadIdx.x;
  if (idx >= total) return;
  int c = idx % C;
  int b = idx / (S * C);
  int bc = b * C + c;
  float v = (in[idx] - stats[2 * bc]) * stats[2 * bc + 1];
  if (mode == 1) v += skip[idx];
  out[idx] = lrelu_f(v);
}

// ---------------------------------------------------------------------------
// conv 3x3x3 SAME, no bias, implicit GEMM via f32 WMMA.
// One wave computes a 48(oc) x 16(voxel) tile: the activation fragment is
// loaded once per K-step and feeds 3 wmmas (3x arithmetic intensity).
// K = 27*C (C % 4 == 0).  Weights in original (OC, IC, 3, 3, 3) layout.
// ---------------------------------------------------------------------------
__global__ void conv_gemm_kernel(const float* __restrict__ in, const float* __restrict__ w,
                                 float* __restrict__ out, int dd, int C, int OC) {
  int lane = threadIdx.x, l15 = lane & 15, hi = lane >> 4;
  int nbase = blockIdx.x * 16, mbase = blockIdx.y * 48;
  int t = nbase + l15;
  int sp = dd * dd * dd;
  int b = t / sp, r0 = t % sp;
  int z = r0 / (dd * dd), y = (r0 / dd) % dd, x = r0 % dd;
  const float* w0 = w + (size_t)(mbase + l15) * C * 27;
  const float* w1 = w0 + (size_t)16 * C * 27;
  const float* w2 = w1 + (size_t)16 * C * 27;
  v8f acc0 = {}, acc1 = {}, acc2 = {};
  for (int kp = 0; kp < 27; kp++) {
    int zz = z + kp / 9 - 1, yy = y + (kp / 3) % 3 - 1, xx = x + kp % 3 - 1;
    bool vmask = (zz >= 0) && (zz < dd) && (yy >= 0) && (yy < dd) && (xx >= 0) && (xx < dd);
    size_t soff = (((size_t)(b * dd + zz) * dd + yy) * dd + xx) * C;
    for (int ic = 0; ic < C; ic += 4) {
      int i0 = ic + hi * 2;
      v2f bf, a0, a1, a2;
      bf.x = vmask ? in[soff + i0] : 0.f;
      bf.y = vmask ? in[soff + i0 + 1] : 0.f;
      a0.x = w0[(size_t)i0 * 27 + kp];      a0.y = w0[(size_t)(i0 + 1) * 27 + kp];
      a1.x = w1[(size_t)i0 * 27 + kp];      a1.y = w1[(size_t)(i0 + 1) * 27 + kp];
      a2.x = w2[(size_t)i0 * 27 + kp];      a2.y = w2[(size_t)(i0 + 1) * 27 + kp];
      acc0 = wmma4(a0, bf, acc0);
      acc1 = wmma4(a1, bf, acc1);
      acc2 = wmma4(a2, bf, acc2);
    }
  }
#pragma unroll
  for (int rr = 0; rr < 8; rr++) {
    int mm = mbase + rr + hi * 8;
    out[(size_t)t * OC + mm] = acc0[rr];
    out[(size_t)t * OC + mm + 16] = acc1[rr];
    out[(size_t)t * OC + mm + 32] = acc2[rr];
  }
}

// ---------------------------------------------------------------------------
// Generic token GEMM: Y[t, m] = act(W[m,:] . X[t,:] + bias[m])
// 48(M) x 16(token) tile per wave; mode 0: store; 1: gelu store; 2: accumulate.
// ntok % 16 == 0, K % 4 == 0, M % 48 == 0.
// ---------------------------------------------------------------------------
__global__ void gemm_tok_kernel(const float* __restrict__ W, const float* __restrict__ bias,
                                const float* __restrict__ Xm, float* __restrict__ Y,
                                int M, int K, int mode) {
  int lane = threadIdx.x, l15 = lane & 15, hi = lane >> 4;
  int nbase = blockIdx.x * 16, mbase = blockIdx.y * 48;
  int t = nbase + l15;
  const float* src = Xm + (size_t)t * K;
  const float* w0 = W + (size_t)(mbase + l15) * K;
  const float* w1 = w0 + (size_t)16 * K;
  const float* w2 = w1 + (size_t)16 * K;
  v8f acc0 = {}, acc1 = {}, acc2 = {};
  for (int kk = 0; kk < K; kk += 4) {
    int i0 = kk + hi * 2;
    v2f b, a0, a1, a2;
    b.x = src[i0]; b.y = src[i0 + 1];
    a0.x = w0[i0]; a0.y = w0[i0 + 1];
    a1.x = w1[i0]; a1.y = w1[i0 + 1];
    a2.x = w2[i0]; a2.y = w2[i0 + 1];
    acc0 = wmma4(a0, b, acc0);
    acc1 = wmma4(a1, b, acc1);
    acc2 = wmma4(a2, b, acc2);
  }
#pragma unroll
  for (int mt = 0; mt < 3; mt++) {
    v8f acc = mt == 0 ? acc0 : (mt == 1 ? acc1 : acc2);
#pragma unroll
    for (int rr = 0; rr < 8; rr++) {
      int mm = mbase + mt * 16 + rr + hi * 8;
      float v = acc[rr] + (bias ? bias[mm] : 0.f);
      size_t di = (size_t)t * M + mm;
      if (mode == 1) Y[di] = gelu_f(v);
      else if (mode == 2) Y[di] += v;
      else Y[di] = v;
    }
  }
}

// ---------------------------------------------------------------------------
// QKV GEMM with fused pad / cyclic-shift / window-partition gather.
// 48(M of 3C) x 16(windowed token) tile per wave.
// Output split into q,k,v with layout (bn, head, i, 16); q pre-scaled by 0.25.
// ---------------------------------------------------------------------------
__global__ void qkv_gemm_kernel(const float* __restrict__ lnx, const float* __restrict__ qw,
                                const float* __restrict__ qb, float* __restrict__ qo,
                                float* __restrict__ ko, float* __restrict__ vo,
                                int Bb, int dd, int C, int H, int wsz, int dp, int nw1,
                                int shifted, int ssh) {
  int lane = threadIdx.x, l15 = lane & 15, hi = lane >> 4;
  int N = wsz * wsz * wsz, nwb = nw1 * nw1 * nw1;
  int NTOK = Bb * nwb * N;
  int nbase = blockIdx.x * 16, mbase = blockIdx.y * 48;
  int t = nbase + l15;
  bool tval = t < NTOK;
  int tc = tval ? t : 0;
  int bn = tc / N, i = tc % N;
  int b = bn / nwb, wnd = bn % nwb;
  int iz = i / (wsz * wsz), iy = (i / wsz) % wsz, ix = i % wsz;
  int wd = wnd / (nw1 * nw1), wh = (wnd / nw1) % nw1, ww = wnd % nw1;
  int pz = wd * wsz + iz, py = wh * wsz + iy, px = ww * wsz + ix;
  if (shifted) { pz = (pz + ssh) % dp; py = (py + ssh) % dp; px = (px + ssh) % dp; }
  bool val = tval && (pz < dd) && (py < dd) && (px < dd);
  size_t soff = (((size_t)(b * dd + pz) * dd + py) * dd + px) * C;
  const float* w0 = qw + (size_t)(mbase + l15) * C;
  const float* w1 = w0 + (size_t)16 * C;
  const float* w2 = w1 + (size_t)16 * C;
  v8f acc0 = {}, acc1 = {}, acc2 = {};
  for (int kk = 0; kk < C; kk += 4) {
    int i0 = kk + hi * 2;
    v2f bf, a0, a1, a2;
    bf.x = val ? lnx[soff + i0] : 0.f;
    bf.y = val ? lnx[soff + i0 + 1] : 0.f;
    a0.x = w0[i0]; a0.y = w0[i0 + 1];
    a1.x = w1[i0]; a1.y = w1[i0 + 1];
    a2.x = w2[i0]; a2.y = w2[i0 + 1];
    acc0 = wmma4(a0, bf, acc0);
    acc1 = wmma4(a1, bf, acc1);
    acc2 = wmma4(a2, bf, acc2);
  }
  if (!tval) return;
#pragma unroll
  for (int mt = 0; mt < 3; mt++) {
    v8f acc = mt == 0 ? acc0 : (mt == 1 ? acc1 : acc2);
#pragma unroll
    for (int rr = 0; rr < 8; rr++) {
      int mm = mbase + mt * 16 + rr + hi * 8;
      float v = acc[rr] + qb[mm];
      int which = mm / C, head = (mm % C) / 16, comp = mm % 16;
      size_t di = (((size_t)bn * H + head) * N + i) * 16 + comp;
      if (which == 0) qo[di] = v * 0.25f;      // hd^-0.5 == 16^-0.5
      else if (which == 1) ko[di] = v;
      else vo[di] = v;
    }
  }
}

// ---------------------------------------------------------------------------
// Fused window attention per (window-batch, head, 16-query tile):
//   - V tile (N x 16) is staged into LDS with GLOBAL_LOAD_ASYNC_TO_LDS_B128
//     (ASYNCcnt), overlapping the whole S-phase WMMAs and the softmax.
//   - S = qK^T + rpb[RPI] (+ shift mask), softmax rows in LDS, O = P.V (V
//     served from LDS after s_wait_asynccnt 0).
//   - Q fragments hoisted once (invariant across all key tiles).
//   - RPI always decoded with the (7,7,7) window, as in the reference.
// ---------------------------------------------------------------------------
__global__ void attn_kernel(const float* __restrict__ qo, const float* __restrict__ ko,
                            const float* __restrict__ vo, const float* __restrict__ rpb,
                            float* __restrict__ o, int H, int N, int C, int wsz,
                            int dp, int nw1, int shifted, int NWB) {
  const int PS = 353;                     // 22*16 = 352 cols max, +1 pad
  __shared__ float P[16 * PS];
  __shared__ float Vs[352 * 16];          // V tile staged via async-to-LDS
  int lane = threadIdx.x, l15 = lane & 15, hi = lane >> 4;
  int bn = blockIdx.z, h = blockIdx.y, qbase = blockIdx.x * 16;
  int NT = (N + 15) / 16;
  size_t base = ((size_t)bn * H + h) * N * 16;
  int wnd = bn % NWB;
  int wd = wnd / (nw1 * nw1), wh = (wnd / nw1) % nw1, ww = wnd % nw1;

  // ---- kick off async copy of V (NT*16 rows x 16 comps, 16B chunks) ----
  // NT*64 chunks is a multiple of 32 -> uniform loop, full EXEC.
  {
    unsigned lbase = (unsigned)(size_t)&Vs[0];
    uint64_t gbase = (uint64_t)(uintptr_t)(vo + base);
    int nit = (NT * 64) / 32;
    for (int it2 = 0; it2 < nit; it2++) {
      unsigned la = lbase + (unsigned)(it2 * 32 + lane) * 16u;
      uint64_t ga = gbase + (uint64_t)(it2 * 32 + lane) * 16u;
      asm volatile("global_load_async_to_lds_b128 %0, %1, off"
                   :: "v"(la), "v"(ga) : "memory");
    }
  }

  // ---- hoist Q fragments (constant across key tiles) ----
  int qi = qbase + l15; if (qi >= N) qi = N - 1;
  v2f aq[4];
#pragma unroll
  for (int k4 = 0; k4 < 4; k4++) {
    int c0 = k4 * 4 + hi * 2;
    aq[k4].x = qo[base + (size_t)qi * 16 + c0];
    aq[k4].y = qo[base + (size_t)qi * 16 + c0 + 1];
  }

  // ---- S tiles -> LDS (overlaps the async V copy) ----
  for (int jt = 0; jt < NT; jt++) {
    int kj = jt * 16 + l15; if (kj >= N) kj = N - 1;
    v8f acc = {};
#pragma unroll
    for (int k4 = 0; k4 < 4; k4++) {
      int c0 = k4 * 4 + hi * 2;
      v2f bf;
      bf.x = ko[base + (size_t)kj * 16 + c0];
      bf.y = ko[base + (size_t)kj * 16 + c0 + 1];
      acc = wmma4(aq[k4], bf, acc);
    }
#pragma unroll
    for (int rr = 0; rr < 8; rr++) {
      int i = qbase + rr + hi * 8;
      int j = jt * 16 + l15;
      float v;
      if (i >= N || j >= N) v = -1e30f;
      else {
        int i7z = i / 49, i7y = (i / 7) % 7, i7x = i % 7;
        int j7z = j / 49, j7y = (j / 7) % 7, j7x = j % 7;
        int bi = (i7z - j7z + 6) * 169 + (i7y - j7y + 6) * 13 + (i7x - j7x + 6);
        v = acc[rr] + rpb[(size_t)bi * H + h];
        if (shifted) {
          int iz = i / (wsz * wsz), iy = (i / wsz) % wsz, ix = i % wsz;
          int jz = j / (wsz * wsz), jy = (j / wsz) % wsz, jx = j % wsz;
          int li = regio7(wd * wsz + iz, dp) * 9 + regio7(wh * wsz + iy, dp) * 3 +
                   regio7(ww * wsz + ix, dp);
          int lj = regio7(wd * wsz + jz, dp) * 9 + regio7(wh * wsz + jy, dp) * 3 +
                   regio7(ww * wsz + jx, dp);
          if (li != lj) v -= 100.f;
        }
      }
      P[(rr + hi * 8) * PS + j] = v;
    }
  }
  __syncthreads();

  // ---- softmax rows (async V copy still in flight underneath) ----
  if (lane < 16) {
    float m = -1e30f;
    for (int j = 0; j < N; j++) m = fmaxf(m, P[lane * PS + j]);
    float s = 0.f;
    for (int j = 0; j < N; j++) {
      float e = __expf(P[lane * PS + j] - m);
      P[lane * PS + j] = e;
      s += e;
    }
    float inv = 1.f / s;
    int NP = NT * 16;
    for (int j = 0; j < NP; j++) P[lane * PS + j] = (j < N) ? P[lane * PS + j] * inv : 0.f;
  }
  __syncthreads();

  // ---- drain async V copy, then O = P . V from LDS ----
  asm volatile("s_wait_asynccnt 0x0" ::: "memory");
  __syncthreads();
  v8f acc = {};
  for (int kt = 0; kt < NT; kt++) {
#pragma unroll
    for (int kk = 0; kk < 16; kk += 4) {
      int krow = kt * 16 + kk + hi * 2;
      v2f a, bf;
      a.x = P[l15 * PS + krow];
      a.y = P[l15 * PS + krow + 1];
      bf.x = Vs[krow * 16 + l15];          // padded rows hit P == 0
      bf.y = Vs[(krow + 1) * 16 + l15];
      acc = wmma4(a, bf, acc);
    }
  }
#pragma unroll
  for (int rr = 0; rr < 8; rr++) {
    int i = qbase + rr + hi * 8;
    if (i < N) o[((size_t)bn * N + i) * C + h * 16 + l15] = acc[rr];
  }
}

// ---------------------------------------------------------------------------
// Output projection GEMM (48 x 16 tile) with fused window-reverse / un-shift /
// crop scatter, residual-added into x (each valid voxel gets exactly one token).
// ---------------------------------------------------------------------------
__global__ void proj_gemm_kernel(const float* __restrict__ o, const float* __restrict__ pw,
                                 const float* __restrict__ pb, float* __restrict__ xout,
                                 int Bb, int dd, int C, int wsz, int dp, int nw1,
                                 int shifted, int ssh) {
  int lane = threadIdx.x, l15 = lane & 15, hi = lane >> 4;
  int N = wsz * wsz * wsz, nwb = nw1 * nw1 * nw1;
  int NTOK = Bb * nwb * N;
  int nbase = blockIdx.x * 16, mbase = blockIdx.y * 48;
  int t = nbase + l15;
  bool tval = t < NTOK;
  int tc = tval ? t : 0;
  int bn = tc / N, i = tc % N;
  int b = bn / nwb, wnd = bn % nwb;
  int iz = i / (wsz * wsz), iy = (i / wsz) % wsz, ix = i % wsz;
  int wd = wnd / (nw1 * nw1), wh = (wnd / nw1) % nw1, ww = wnd % nw1;
  int pz = wd * wsz + iz, py = wh * wsz + iy, px = ww * wsz + ix;
  if (shifted) { pz = (pz + ssh) % dp; py = (py + ssh) % dp; px = (px + ssh) % dp; }
  bool val = tval && (pz < dd) && (py < dd) && (px < dd);
  const float* src = o + (size_t)tc * C;
  const float* w0 = pw + (size_t)(mbase + l15) * C;
  const float* w1 = w0 + (size_t)16 * C;
  const float* w2 = w1 + (size_t)16 * C;
  v8f acc0 = {}, acc1 = {}, acc2 = {};
  for (int kk = 0; kk < C; kk += 4) {
    int i0 = kk + hi * 2;
    v2f bf, a0, a1, a2;
    bf.x = src[i0]; bf.y = src[i0 + 1];
    a0.x = w0[i0]; a0.y = w0[i0 + 1];
    a1.x = w1[i0]; a1.y = w1[i0 + 1];
    a2.x = w2[i0]; a2.y = w2[i0 + 1];
    acc0 = wmma4(a0, bf, acc0);
    acc1 = wmma4(a1, bf, acc1);
    acc2 = wmma4(a2, bf, acc2);
  }
  if (val) {
    float* dst = xout + (((size_t)(b * dd + pz) * dd + py) * dd + px) * C;
#pragma unroll
    for (int rr = 0; rr < 8; rr++) {
      int mm = mbase + rr + hi * 8;
      dst[mm] += acc0[rr] + pb[mm];
      dst[mm + 16] += acc1[rr] + pb[mm + 16];
      dst[mm + 32] += acc2[rr] + pb[mm + 32];
    }
  }
}

// ---------------------------------------------------------------------------
// PatchMerging gather (with MONAI's duplicated offsets) + affine LN over 8C
// ---------------------------------------------------------------------------
__global__ void merge_ln_kernel(const float* __restrict__ xin, float* __restrict__ out8,
                                const float* __restrict__ g, const float* __restrict__ bb,
                                int dd, int C, int nvox2) {
  const int offz[8] = {0, 1, 0, 0, 1, 0, 0, 1};
  const int offy[8] = {0, 0, 1, 0, 0, 1, 0, 1};
  const int offx[8] = {0, 0, 0, 1, 1, 0, 1, 1};
  int wid = (blockIdx.x * blockDim.x + threadIdx.x) >> 5;
  int lane = threadIdx.x & 31;
  if (wid >= nvox2) return;
  int d2 = dd / 2, sp2 = d2 * d2 * d2;
  int b = wid / sp2, r0 = wid % sp2;
  int z2 = r0 / (d2 * d2), y2 = (r0 / d2) % d2, x2 = r0 % d2;
  int C8 = 8 * C;
  float s = 0.f, s2 = 0.f;
  for (int f = lane; f < C8; f += 32) {
    int sl = f / C, c = f % C;
    float v = xin[(((size_t)(b * dd + 2 * z2 + offz[sl]) * dd + 2 * y2 + offy[sl]) * dd +
                   2 * x2 + offx[sl]) * C + c];
    s += v; s2 += v * v;
  }
  s = wred(s); s2 = wred(s2);
  float m = s / C8;
  float r = rsqrtf(s2 / C8 - m * m + EPSN);
  float* q = out8 + (size_t)wid * C8;
  for (int f = lane; f < C8; f += 32) {
    int sl = f / C, c = f % C;
    float v = xin[(((size_t)(b * dd + 2 * z2 + offz[sl]) * dd + 2 * y2 + offy[sl]) * dd +
                   2 * x2 + offx[sl]) * C + c];
    q[f] = (v - m) * r * g[f] + bb[f];
  }
}

// ---------------------------------------------------------------------------
// Host orchestration
// ---------------------------------------------------------------------------
extern "C" void kernel_launch(void* const* d_in, const int* in_sizes, int n_in,
                              void* d_out, int out_size, void* d_ws, size_t ws_size,
                              hipStream_t stream) {
  const int B = 2;
  // input "x" is either last (jax sorted-key pytree) or first (insertion order)
  int xi = n_in - 1, base = 0;
  if (n_in > 0 && in_sizes[0] == B * 64 * 64 * 64) { xi = 0; base = 1; }
  const float* X = (const float*)d_in[xi];
  const float* pe_b = (const float*)d_in[base + 0];
  const float* pe_w = (const float*)d_in[base + 1];

  float* ws = (float*)d_ws;
  const size_t CAPX = 3200000;  // max token tensor: 2*32^3*48 = 3,145,728 floats
  float* bx = ws;               // current feature map, channels-last tokens
  float* bln = bx + CAPX;       // LN output / conv t1 / merge-8C buffer
  float* bt = bln + CAPX;       // merge output (<= 786432)
  float* bo = bt + 1000000;     // attn windowed output / conv t2 (<= 4,116,000)
  float* bh = bo + 4200000;     // q|k|v (3 x <=4,116,000)  OR  mlp hidden (<=12,582,912)
  float* bst = bh + 12600000;   // instance-norm stats
  size_t need_bytes = (size_t)(bst + 4096 - ws) * sizeof(float);
  if (ws_size < need_bytes) return;

  float* outp = (float*)d_out;
  const int HEADSv[4] = {3, 6, 12, 24};

  // ---- patch embed (2,1,64,64,64) -> (2,32^3,48) channels-last ----
  {
    int total = B * 32 * 32 * 32 * 48;
    pe_kernel<<<dim3((total + 255) / 256), dim3(256), 0, stream>>>(X, pe_w, pe_b, bx, total);
  }
  size_t ooff = 0;
  {
    int nvox = B * 32 * 32 * 32;
    proj_out_kernel<<<dim3((nvox + 3) / 4), dim3(128), 0, stream>>>(bx, outp, nvox, 48,
                                                                    32 * 32 * 32);
    ooff += (size_t)B * 48 * 32 * 32 * 32;
  }

  int C = 48, dd = 32;
  for (int s = 0; s < 4; s++) {
    int sb = base + 2 + 31 * s;  // 13+13 block leaves + 2 conv + 3 merge
    const float* conv1_w = (const float*)d_in[sb + 26];
    const float* conv2_w = (const float*)d_in[sb + 27];
    const float* mnorm_b = (const float*)d_in[sb + 28];
    const float* mnorm_g = (const float*)d_in[sb + 29];
    const float* red_w = (const float*)d_in[sb + 30];

    int S = dd * dd * dd;
    int ntok = B * S;
    int total = ntok * C;

    // ---- UnetResBlock (use_v2 pre-stage conv) ----
    conv_gemm_kernel<<<dim3(ntok / 16, C / 48), dim3(32), 0, stream>>>(bx, conv1_w, bln, dd, C, C);
    inorm_stats_kernel<<<dim3(B * C), dim3(256), 0, stream>>>(bln, bst, C, S);
    inorm_apply_kernel<<<dim3((total + 255) / 256), dim3(256), 0, stream>>>(bln, bst, bln, bln,
                                                                            total, C, S, 0);
    conv_gemm_kernel<<<dim3(ntok / 16, C / 48), dim3(32), 0, stream>>>(bln, conv2_w, bo, dd, C, C);
    inorm_stats_kernel<<<dim3(B * C), dim3(256), 0, stream>>>(bo, bst, C, S);
    inorm_apply_kernel<<<dim3((total + 255) / 256), dim3(256), 0, stream>>>(bo, bst, bx, bx,
                                                                            total, C, S, 1);

    // ---- window geometry ----
    int wsz = dd > 7 ? 7 : dd;
    int ssh = dd > 7 ? 3 : 0;
    int dp = ((dd + wsz - 1) / wsz) * wsz;
    int nw1 = dp / wsz;
    int N = wsz * wsz * wsz;
    int NWB = nw1 * nw1 * nw1;
    int NW = B * NWB;
    int NTOKW = NW * N;
    int H = HEADSv[s];
    float* qb_ = bh;
    float* kb_ = bh + 4200000;
    float* vb_ = bh + 8400000;

    for (int j = 0; j < 2; j++) {
      const float* fc1_b = (const float*)d_in[sb + 13 * j + 0];
      const float* fc1_w = (const float*)d_in[sb + 13 * j + 1];
      const float* fc2_b = (const float*)d_in[sb + 13 * j + 2];
      const float* fc2_w = (const float*)d_in[sb + 13 * j + 3];
      const float* n1b = (const float*)d_in[sb + 13 * j + 4];
      const float* n1g = (const float*)d_in[sb + 13 * j + 5];
      const float* n2b = (const float*)d_in[sb + 13 * j + 6];
      const float* n2g = (const float*)d_in[sb + 13 * j + 7];
      const float* pjb = (const float*)d_in[sb + 13 * j + 8];
      const float* pjw = (const float*)d_in[sb + 13 * j + 9];
      const float* qkvb = (const float*)d_in[sb + 13 * j + 10];
      const float* qkvw = (const float*)d_in[sb + 13 * j + 11];
      const float* rpb = (const float*)d_in[sb + 13 * j + 12];
      int shifted = (j == 1 && ssh > 0) ? 1 : 0;

      ln_kernel<<<dim3((ntok + 3) / 4), dim3(128), 0, stream>>>(bx, bln, n1g, n1b, ntok, C);
      qkv_gemm_kernel<<<dim3((NTOKW + 15) / 16, (3 * C) / 48), dim3(32), 0, stream>>>(
          bln, qkvw, qkvb, qb_, kb_, vb_, B, dd, C, H, wsz, dp, nw1, shifted, ssh);
      attn_kernel<<<dim3((N + 15) / 16, H, NW), dim3(32), 0, stream>>>(
          qb_, kb_, vb_, rpb, bo, H, N, C, wsz, dp, nw1, shifted, NWB);
      proj_gemm_kernel<<<dim3((NTOKW + 15) / 16, C / 48), dim3(32), 0, stream>>>(
          bo, pjw, pjb, bx, B, dd, C, wsz, dp, nw1, shifted, ssh);
      ln_kernel<<<dim3((ntok + 3) / 4), dim3(128), 0, stream>>>(bx, bln, n2g, n2b, ntok, C);
      gemm_tok_kernel<<<dim3(ntok / 16, (4 * C) / 48), dim3(32), 0, stream>>>(
          fc1_w, fc1_b, bln, bh, 4 * C, C, 1);
      gemm_tok_kernel<<<dim3(ntok / 16, C / 48), dim3(32), 0, stream>>>(
          fc2_w, fc2_b, bh, bx, C, 4 * C, 2);
    }

    // ---- PatchMerging ----
    int d2 = dd / 2;
    int nvox2 = B * d2 * d2 * d2;
    merge_ln_kernel<<<dim3((nvox2 + 3) / 4), dim3(128), 0, stream>>>(bx, bln, mnorm_g, mnorm_b,
                                                                     dd, C, nvox2);
    gemm_tok_kernel<<<dim3(nvox2 / 16, (2 * C) / 48), dim3(32), 0, stream>>>(
        red_w, nullptr, bln, bt, 2 * C, 8 * C, 0);
    hipMemcpyAsync(bx, bt, (size_t)nvox2 * 2 * C * sizeof(float),
                   hipMemcpyDeviceToDevice, stream);
    C *= 2;
    dd = d2;
    int nvox = B * dd * dd * dd;
    proj_out_kernel<<<dim3((nvox + 3) / 4), dim3(128), 0, stream>>>(bx, outp + ooff, nvox, C,
                                                                    dd * dd * dd);
    ooff += (size_t)nvox * C;
  }
  (void)out_size;
}